// e3nn_tp_42090679501107
// MI455X (gfx1250) — compile-verified
//
#include <hip/hip_runtime.h>

typedef float v2f __attribute__((ext_vector_type(2)));
typedef float v8f __attribute__((ext_vector_type(8)));

// ---------------------------------------------------------------------------
// Prep kernel: build combined weight matrix Wcat[16][48] (fp32) in d_ws.
// out[k] = sum_q Wcat[k][q] * v[q],  v[q] = a[q&15] * b[q>>4]
// Mirrors the reference wigner_3j construction exactly (fp64 on device).
// ---------------------------------------------------------------------------

struct cx { double re, im; };
__device__ inline cx cmul(cx a, cx b){ return {a.re*b.re - a.im*b.im, a.re*b.im + a.im*b.re}; }
__device__ inline cx cconj(cx a){ return {a.re, -a.im}; }

__device__ double factd(int n){ double r = 1.0; for (int i = 2; i <= n; ++i) r *= (double)i; return r; }

__device__ double su2_cg(int j1,int m1,int j2,int m2,int j3,int m3){
  if (m3 != m1 + m2) return 0.0;
  int vmin = -j1 + j2 + m3; if (-j1 + m1 > vmin) vmin = -j1 + m1; if (vmin < 0) vmin = 0;
  int vmax = j2 + j3 + m1; if (j3 - j1 + j2 < vmax) vmax = j3 - j1 + j2; if (j3 + m3 < vmax) vmax = j3 + m3;
  double c = sqrt((2.0*j3 + 1.0) * factd(j3+j1-j2) * factd(j3-j1+j2) * factd(j1+j2-j3)
                  * factd(j3+m3) * factd(j3-m3)
                  / (factd(j1+j2+j3+1) * factd(j1-m1) * factd(j1+m1) * factd(j2-m2) * factd(j2+m2)));
  double s = 0.0;
  for (int v = vmin; v <= vmax; ++v){
    double sgn = ((v + j2 + m2) & 1) ? -1.0 : 1.0;
    s += sgn / factd(v) * factd(j2+j3+m1-v) * factd(j1-m1+v)
         / (factd(j3-j1+j2-v) * factd(j3+m3-v) * factd(v+j1-j2-m3));
  }
  return c * s;
}

__device__ void real_from_complex(int l, cx Q[7][7]){
  for (int i = 0; i < 7; ++i) for (int j = 0; j < 7; ++j) Q[i][j] = {0.0, 0.0};
  const double s2 = 0.70710678118654752440;
  for (int m = -l; m < 0; ++m){
    Q[l+m][l-m] = { s2, 0.0};
    Q[l+m][l+m] = { 0.0, -s2};
  }
  Q[l][l] = {1.0, 0.0};
  for (int m = 1; m <= l; ++m){
    double sg = (m & 1) ? -1.0 : 1.0;
    Q[l+m][l+m] = { sg*s2, 0.0 };
    Q[l+m][l-m] = { 0.0, sg*s2 };
  }
  cx ph;                       // (-i)^l
  switch (l & 3){
    case 0:  ph = { 1.0, 0.0}; break;
    case 1:  ph = { 0.0,-1.0}; break;
    case 2:  ph = {-1.0, 0.0}; break;
    default: ph = { 0.0, 1.0}; break;
  }
  for (int i = 0; i < 7; ++i) for (int j = 0; j < 7; ++j) Q[i][j] = cmul(ph, Q[i][j]);
}

__global__ void e3nn_tp_prep(const float* __restrict__ weights, float* __restrict__ wc){
  if (threadIdx.x != 0 || blockIdx.x != 0) return;
  const int path_l1[9] = {0,1,1,1,2,2,2,3,3};
  const int path_lo[9] = {1,0,1,2,1,2,3,2,3};
  const double fan[4]  = {1.0, 3.0, 3.0, 2.0};

  double Wcat[16][48];
  for (int k = 0; k < 16; ++k) for (int q = 0; q < 48; ++q) Wcat[k][q] = 0.0;

  for (int p = 0; p < 9; ++p){
    const int l1 = path_l1[p], lo = path_lo[p];
    const int d1 = 2*l1 + 1, d3 = 2*lo + 1;

    double Ccg[7][3][7];
    for (int i = 0; i < 7; ++i) for (int j = 0; j < 3; ++j) for (int k = 0; k < 7; ++k) Ccg[i][j][k] = 0.0;
    for (int m1 = -l1; m1 <= l1; ++m1)
      for (int m2 = -1; m2 <= 1; ++m2){
        int m3 = m1 + m2;
        if (m3 >= -lo && m3 <= lo)
          Ccg[l1+m1][1+m2][lo+m3] = su2_cg(l1, m1, 1, m2, lo, m3);
      }

    cx Q1[7][7], Q2[7][7], Q3[7][7];
    real_from_complex(l1, Q1);
    real_from_complex(1,  Q2);
    real_from_complex(lo, Q3);

    double Cr[7][3][7];
    double norm2 = 0.0;
    for (int j = 0; j < d1; ++j)
      for (int l = 0; l < 3; ++l)
        for (int n = 0; n < d3; ++n){
          cx acc = {0.0, 0.0};
          for (int i = 0; i < d1; ++i){
            if (Q1[i][j].re == 0.0 && Q1[i][j].im == 0.0) continue;
            for (int k = 0; k < 3; ++k){
              if (Q2[k][l].re == 0.0 && Q2[k][l].im == 0.0) continue;
              for (int m = 0; m < d3; ++m){
                double cg = Ccg[i][k][m];
                if (cg == 0.0) continue;
                cx t = cmul(Q1[i][j], Q2[k][l]);
                t = cmul(t, cconj(Q3[m][n]));
                acc.re += t.re * cg;
                acc.im += t.im * cg;
              }
            }
          }
          Cr[j][l][n] = acc.re;
          norm2 += acc.re * acc.re;
        }

    double inv = (norm2 > 0.0) ? 1.0 / sqrt(norm2) : 0.0;
    double pw  = sqrt((2.0*lo + 1.0) / fan[lo]) * (double)weights[p] * inv;
    for (int i = 0; i < d1; ++i)
      for (int j = 0; j < 3; ++j)
        for (int k = 0; k < d3; ++k)
          Wcat[lo*lo + k][j*16 + l1*l1 + i] += pw * Cr[i][j][k];
  }

  for (int k = 0; k < 16; ++k)
    for (int q = 0; q < 48; ++q)
      wc[k*48 + q] = (float)Wcat[k][q];
}

// ---------------------------------------------------------------------------
// Main kernel: out(16 x S) = Wcat(16 x 48) x V(48 x S), V[q][s]=a_i[s]*b_j[s]
// One wave per row n; 4 sample-tiles of 16 over C=64; 12 chained
// v_wmma_f32_16x16x4_f32 per tile.
//   A layout (16x4 f32): lane(h,l): M=l, VGPR0=K(2h), VGPR1=K(2h+1)
//   B layout (4x16 f32): lane(h,l): N=l, VGPR0=K(2h), VGPR1=K(2h+1)
//   D layout (16x16 f32): VGPR r: lanes0-15 M=r, lanes16-31 M=r+8
// Lane (h,l) only ever touches K-indices q = 4w+2h, 4w+2h+1, i.e. the 8
// x1 rows i = 4u+2h, 4u+2h+1 (u=0..3) of its sample -- we fold the h
// dependence into the load base pointer so every register index below is a
// compile-time constant (no v_cndmask select trees).
// ---------------------------------------------------------------------------
__global__ __launch_bounds__(256) void e3nn_tp_main(const float* __restrict__ x1,
                                                    const float* __restrict__ x2,
                                                    const float* __restrict__ wc,
                                                    float* __restrict__ out, int N)
{
  const int lane = threadIdx.x & 31;
  const int wave = threadIdx.x >> 5;
  const int n    = blockIdx.x * 8 + wave;
  if (n >= N) return;                 // wave-uniform: EXEC stays all-ones

  const int l = lane & 15;
  const int h = lane >> 4;

  // Load A fragments for all 12 WMMAs (constant for whole kernel)
  v2f A[12];
  const float* wrow = wc + l * 48 + 2 * h;
#pragma unroll
  for (int w = 0; w < 12; ++w)
    A[w] = *(const v2f*)(wrow + 4 * w);

  const float* x1p = x1 + (size_t)n * 1024;   // (16, 64)
  const float* x2p = x2 + (size_t)n * 192;    // (3, 64)
  float*       op  = out + (size_t)n * 1024;  // (16, 64)

  // h-dependent bases: lane (h,l) reads x1 rows {4u+2h, 4u+2h+1} and writes
  // out rows {r+8h}.
  const float* x1b = x1p + 2 * h * 64;        // + row offset 2h
  float*       ob  = op + 8 * h * 64;         // + row offset 8h

  // Prefetch the rows the next block's corresponding wave will read
  __builtin_prefetch(x1p + 8 * 1024, 0, 0);
  __builtin_prefetch(x2p + 8 * 192, 0, 0);

#pragma unroll
  for (int t = 0; t < 4; ++t){
    const int c = t * 16 + l;

    // a_vals[m] = x1[n, 4*(m>>1) + 2h + (m&1), c]  -- constant immediate
    // offsets off the h-folded base pointer.
    float av[8];
    const float* ax = x1b + c;
#pragma unroll
    for (int m = 0; m < 8; ++m)
      av[m] = ax[(4 * (m >> 1) + (m & 1)) * 64];

    float b[3];
#pragma unroll
    for (int j = 0; j < 3; ++j) b[j] = x2p[j * 64 + c];

    v8f D = {0.f, 0.f, 0.f, 0.f, 0.f, 0.f, 0.f, 0.f};
#pragma unroll
    for (int w = 0; w < 12; ++w){
      const int j = w >> 2;               // which b component
      const int u = w & 3;                // which a pair
      v2f B;
      B.x = av[2 * u]     * b[j];
      B.y = av[2 * u + 1] * b[j];
      D = __builtin_amdgcn_wmma_f32_16x16x4_f32(
            /*neg_a=*/false, A[w], /*neg_b=*/false, B,
            /*c_mod=*/(short)0, D, /*reuse_a=*/false, /*reuse_b=*/false);
    }

#pragma unroll
    for (int r = 0; r < 8; ++r)
      ob[r * 64 + c] = D[r];
  }
}

// ---------------------------------------------------------------------------
extern "C" void kernel_launch(void* const* d_in, const int* in_sizes, int n_in,
                              void* d_out, int out_size, void* d_ws, size_t ws_size,
                              hipStream_t stream) {
  const float* x1      = (const float*)d_in[0];   // (N, 16, 64)
  const float* x2      = (const float*)d_in[1];   // (N, 3, 64)
  const float* weights = (const float*)d_in[2];   // (9,)
  float* out = (float*)d_out;                     // (N, 16, 64)
  float* wc  = (float*)d_ws;                      // Wcat[16][48]

  const int N = in_sizes[0] / (16 * 64);          // 50000

  e3nn_tp_prep<<<1, 32, 0, stream>>>(weights, wc);

  const int blocks = (N + 7) / 8;                 // 8 waves (rows) per block
  e3nn_tp_main<<<blocks, 256, 0, stream>>>(x1, x2, wc, out, N);
}